// DelayedXOR_SH_SNN_Improved_2637109920462
// MI455X (gfx1250) — compile-verified
//
#include <hip/hip_runtime.h>
#include <math.h>

// MI455X / gfx1250, wave32. One wave per batch element, lane = hidden unit h.
//
// Input projection via V_WMMA_F32_16X16X4_F32 with BOTH groups folded into K:
//   A rows 0..7  : x[t0+m]      in K=0,1  (K=2,3 = 0)
//   A rows 8..15 : x[t0+(m-8)]  in K=2,3  (K=0,1 = 0)
//   B k=0,1 = W_group0 ; k=2,3 = W_group1
// => D rows 0..7 = group0 gi, rows 8..15 = group1 gi, and the fixed D layout
//    (rows 0..7 -> lanes 0..15, rows 8..15 -> lanes 16..31) puts gi[t0+r, h]
//    exactly in lane h, VGPR r. No cross-lane shuffles, no routing selects.

typedef __attribute__((ext_vector_type(2))) float v2f;   // A/B operand: 2 VGPRs
typedef __attribute__((ext_vector_type(8))) float v8f;   // C/D operand: 8 VGPRs

#define SEQ_T            1024
#define CHUNK            32            // timesteps per outer iteration (4 WMMA tiles)
#define NCHUNK           (SEQ_T / CHUNK)
#define ACC_START_CHUNK  24            // decision window starts at t = 768 = 24*32

__device__ __forceinline__ void lif_step(float gi,
                                         float alpha, float beta,
                                         float alphaS, float betaS,
                                         float& v, float& s,
                                         float& V, float& S,
                                         float& ssum) {
  v = fmaf(alpha, v, fmaf(beta, gi, -s));      // alpha*v + (1-alpha)*gi - s
  s = (v > 1.0f) ? 1.0f : 0.0f;                // spike(v - V_TH)
  V = fmaf(alphaS, V, fmaf(betaS, s, -S));
  S = (V > 1.0f) ? 1.0f : 0.0f;
  ssum += S;
}

__global__ __launch_bounds__(128, 1)
void snn_scan_kernel(const float* __restrict__ x,         // [B, T, 2]
                     const float* __restrict__ Wg,        // [2, 16, 2] -> flat [h][i]
                     const float* __restrict__ tau_m,     // [2, 16]    -> flat [h]
                     const float* __restrict__ soma_tau,  // [32]
                     const float* __restrict__ W_out,     // [1, 32]
                     const float* __restrict__ b_out,     // [1]
                     float* __restrict__ out)             // [B, 1]
{
  const int lane = threadIdx.x & 31;
  const int b    = (blockIdx.x * blockDim.x + threadIdx.x) >> 5;  // wave = batch

  // ---- per-lane (per hidden unit h = lane) constants -------------------------
  const float alpha  = 1.0f / (1.0f + expf(-tau_m[lane]));
  const float beta   = 1.0f - alpha;
  const float alphaS = 1.0f / (1.0f + expf(-soma_tau[lane]));
  const float betaS  = 1.0f - alphaS;
  const float wout   = W_out[lane];

  // ---- B operand: k=0,1 -> group0 weights, k=2,3 -> group1 weights -----------
  // B layout (4x16): lane<16: VGPR0=B[0][n],VGPR1=B[1][n]; lane>=16: B[2],B[3].
  // With B[2g+i][j] = Wg[g][j][i] this is simply Wg[lane*2 + i] for every lane.
  v2f Bw;
  Bw[0] = Wg[lane * 2 + 0];
  Bw[1] = Wg[lane * 2 + 1];

  // A-operand active lanes: rows 0..7 in K=0,1 (lanes 0..7) and rows 8..15 in
  // K=2,3 (lanes 24..31); lanes 8..23 must supply the zero padding.
  const bool act   = (lane < 8) || (lane >= 24);
  const int  tsub  = lane & 7;                  // timestep-within-tile this lane feeds

  const v2f* __restrict__ xb = reinterpret_cast<const v2f*>(x) + (size_t)b * SEQ_T;

  float v = 0.0f, s = 0.0f, V = 0.0f, S = 0.0f, acc = 0.0f;

  // prefetch chunk 0: tile q covers timesteps q*8 .. q*8+7
  v2f xr[4];
  #pragma unroll
  for (int q = 0; q < 4; ++q) xr[q] = xb[q * 8 + tsub];

  for (int c = 0; c < NCHUNK; ++c) {
    // ---- 4 WMMAs: tile q -> gi[c*32 + q*8 + r, h] in lane h, D[q][r] ---------
    const v8f zc = {};
    v8f D[4];
    #pragma unroll
    for (int q = 0; q < 4; ++q) {
      v2f A;
      A[0] = act ? xr[q][0] : 0.0f;
      A[1] = act ? xr[q][1] : 0.0f;
      D[q] = __builtin_amdgcn_wmma_f32_16x16x4_f32(false, A, false, Bw,
                                                   (short)0, zc, false, false);
    }

    // ---- software prefetch of next chunk (overlaps the serial scan below) ----
    const int cn = (c + 1 < NCHUNK) ? (c + 1) : (NCHUNK - 1);
    #pragma unroll
    for (int q = 0; q < 4; ++q) xr[q] = xb[cn * CHUNK + q * 8 + tsub];

    // ---- 32 LIF steps straight out of the D registers ------------------------
    float ssum = 0.0f;
    #pragma unroll
    for (int q = 0; q < 4; ++q) {
      #pragma unroll
      for (int r = 0; r < 8; ++r) {
        lif_step(D[q][r], alpha, beta, alphaS, betaS, v, s, V, S, ssum);
      }
    }

    if (c >= ACC_START_CHUNK) acc += ssum;      // wave-uniform branch
  }

  // ---- output projection: out[b] = sum_h acc[h] * W_out[h] + b_out -----------
  float r = acc * wout;
  #pragma unroll
  for (int off = 16; off > 0; off >>= 1) r += __shfl_xor(r, off);
  if (lane == 0) out[b] = r + b_out[0];
}

extern "C" void kernel_launch(void* const* d_in, const int* in_sizes, int n_in,
                              void* d_out, int out_size, void* d_ws, size_t ws_size,
                              hipStream_t stream) {
  (void)in_sizes; (void)n_in; (void)d_ws; (void)ws_size; (void)out_size;
  const float* x        = (const float*)d_in[0];  // [1024,1024,2]
  const float* Wg       = (const float*)d_in[1];  // [2,16,2]
  const float* tau_m    = (const float*)d_in[2];  // [2,16]
  const float* soma_tau = (const float*)d_in[3];  // [32]
  const float* W_out    = (const float*)d_in[4];  // [1,32]
  const float* b_out    = (const float*)d_in[5];  // [1]
  float* out            = (float*)d_out;          // [1024,1]

  const int threads = 128;                 // 4 waves = 4 batch elements per block
  const int blocks  = (1024 * 32) / threads;
  snn_scan_kernel<<<blocks, threads, 0, stream>>>(x, Wg, tau_m, soma_tau,
                                                  W_out, b_out, out);
}